// MaskedAttentionBlock_53841710023038
// MI455X (gfx1250) — compile-verified
//
#include <hip/hip_runtime.h>
#include <hip/hip_bf16.h>

typedef __attribute__((ext_vector_type(2))) float v2f;
typedef __attribute__((ext_vector_type(8))) float v8f;

#define NPOS   (8 * 64 * 64)     // B*H*W
#define CCH    128               // C == KEY_DIM
#define KS2    25
#define NEGBIG (-1e30f)

#define SWP       288            // W pair-row stride (dwords); 288%64==32 -> disjoint half-wave banks
#define SA_STRIDE 132            // A row stride (floats): 4*l16 bank pattern, conflict-free

// ---------------------------------------------------------------------------
// GEMM: C[M,128] = A[M,128] @ W[128,128] + bias[128], fp32 WMMA 16x16x4.
// Block = 4 waves. W staged once per block into LDS in K-PAIR-INTERLEAVED
// layout sW[kpair*SWP + col*2 + (k&1)] so each B fragment (rows ka, ka+1 at
// one column) is one contiguous ds_load_b64 straight into the WMMA operand
// pair (no repack movs). Each wave's 16x128 A tile staged via coalesced
// float4 loads. One wave computes 16(M) x 128(N): 8 acc tiles, 32 K-steps.
// WMMA fragment layouts per CDNA5 ISA:
//   A: lanes 0-15 -> (M=m0+lane, K=k0+{0,1}), lanes 16-31 -> (M, K=k0+{2,3})
//   B: lanes 0-15 -> (N=n0+lane, K rows k0+{0,1}), lanes 16-31 -> K rows +2
//   D: VGPR r -> row m0 + r + (lane/16)*8, col n0 + (lane&15)
// ---------------------------------------------------------------------------
__global__ __launch_bounds__(128) void gemm_wmma(
    const float* __restrict__ A, const float* __restrict__ W,
    const float* __restrict__ bias, float* __restrict__ Cmat)
{
    __shared__ float sW[64 * SWP];              // 73,728 B (64 K-pairs)
    __shared__ float sA[4][16 * SA_STRIDE];     // 33,792 B

    const int tid  = threadIdx.x;
    const int lane = tid & 31;
    const int wib  = tid >> 5;
    const int m0   = (blockIdx.x * 4 + wib) * 16;

    // Stage W pair-interleaved: 64 pair-rows x 32 col-chunks; 128 threads x 16.
    // Global side: two coalesced float4 row reads; LDS side: two contiguous
    // float4 stores of {r0c0,r1c0,r0c1,r1c1} / {r0c2,r1c2,r0c3,r1c3}.
#pragma unroll
    for (int i = 0; i < 16; ++i) {
        const int p  = i * 128 + tid;       // 0..2047
        const int j  = p >> 5;              // K-pair row 0..63
        const int cc = (p & 31) * 4;        // column chunk
        const float4 g0 = *(const float4*)&W[(size_t)(2 * j)     * CCH + cc];
        const float4 g1 = *(const float4*)&W[(size_t)(2 * j + 1) * CCH + cc];
        const float4 lo = make_float4(g0.x, g1.x, g0.y, g1.y);
        const float4 hi = make_float4(g0.z, g1.z, g0.w, g1.w);
        *(float4*)&sW[j * SWP + cc * 2]     = lo;
        *(float4*)&sW[j * SWP + cc * 2 + 4] = hi;
    }
    // Stage this wave's A tile: 2048 floats, 32 lanes x 16 float4 (coalesced)
#pragma unroll
    for (int i = 0; i < 16; ++i) {
        const int fidx = (i * 32 + lane) * 4;
        const int r = fidx >> 7, c = fidx & 127;
        *(float4*)&sA[wib][r * SA_STRIDE + c] =
            *(const float4*)&A[(size_t)(m0 + r) * CCH + c];
    }
    __syncthreads();

    const int half = lane >> 4;      // 0 or 1
    const int l16  = lane & 15;

    v8f acc[8] = {};

    for (int k0 = 0; k0 < CCH; k0 += 4) {
        const int kp = (k0 >> 1) + half;                 // K-pair row for this half-wave
        const v2f a  = *(const v2f*)&sA[wib][l16 * SA_STRIDE + k0 + half * 2];
#pragma unroll
        for (int t = 0; t < 8; ++t) {
            const int col = t * 16 + l16;
            const v2f b = *(const v2f*)&sW[kp * SWP + col * 2];  // ds_load_b64
            acc[t] = __builtin_amdgcn_wmma_f32_16x16x4_f32(
                false, a, false, b, (short)0, acc[t], false, false);
        }
    }

#pragma unroll
    for (int t = 0; t < 8; ++t) {
        const int col = t * 16 + l16;
        const float bc = bias[col];
#pragma unroll
        for (int r = 0; r < 8; ++r) {
            const int row = m0 + r + half * 8;
            Cmat[(size_t)row * CCH + col] = acc[t][r] + bc;
        }
    }
}

// ---------------------------------------------------------------------------
// 25-tap dilated local attention. One wave per (b,h,w) position; each lane
// owns 4 contiguous channels (float4 -> global_load_b128, fully coalesced).
// Logits via wave32 xor-shuffle reductions; softmax((QK + pm [+ M]) / sqrt(128)).
// OOB taps get -1e30 (== reference pad mask); their softmax weight is exactly 0.
// ---------------------------------------------------------------------------
template <bool HAS_M, bool HAS_V>
__global__ __launch_bounds__(256) void attn_kernel(
    const float* __restrict__ Q, const float* __restrict__ K,
    const float* __restrict__ V, const float* __restrict__ Min,
    float* __restrict__ Out, float* __restrict__ Mout)
{
    const int lane = threadIdx.x & 31;
    const int wid  = blockIdx.x * 8 + (threadIdx.x >> 5);
    if (wid >= NPOS) return;

    const int b  = wid >> 12;
    const int hw = wid & 4095;
    const int h  = hw >> 6;
    const int w  = hw & 63;
    const int c0 = lane * 4;

    const float4 q = *(const float4*)(Q + (size_t)wid * CCH + c0);

    float logit[KS2];
    int   nbr[KS2];

#pragma unroll
    for (int ky = 0; ky < 5; ++ky) {
#pragma unroll
        for (int kx = 0; kx < 5; ++kx) {
            const int t = ky * 5 + kx;
            const int y = h + ky * 2 - 4;
            const int x = w + kx * 2 - 4;
            const bool ok = (y >= 0) && (y < 64) && (x >= 0) && (x < 64); // wave-uniform
            nbr[t] = ok ? ((b * 64 + y) * 64 + x) : -1;
            float p = 0.0f;
            if (ok) {
                const float4 kk = *(const float4*)(K + (size_t)nbr[t] * CCH + c0);
                p = q.x * kk.x + q.y * kk.y + q.z * kk.z + q.w * kk.w;
            }
#pragma unroll
            for (int off = 16; off >= 1; off >>= 1)
                p += __shfl_xor(p, off, 32);
            float lg = ok ? p : NEGBIG;
            if (HAS_M) lg += Min[(size_t)wid * KS2 + t];  // masked taps carry M==0
            logit[t] = lg;
        }
    }

    const float invScale = 0.08838834764831844f;  // 1/sqrt(128)
    float mx = -3.4e38f;
#pragma unroll
    for (int t = 0; t < KS2; ++t) {
        logit[t] *= invScale;
        mx = fmaxf(mx, logit[t]);
    }
    float den = 0.0f;
#pragma unroll
    for (int t = 0; t < KS2; ++t) {
        logit[t] = __expf(logit[t] - mx);
        den += logit[t];
    }
    const float rden = 1.0f / den;
#pragma unroll
    for (int t = 0; t < KS2; ++t) logit[t] *= rden;

    // write scores: lane t stores tap t
#pragma unroll
    for (int t = 0; t < KS2; ++t)
        if (lane == t) Mout[(size_t)wid * KS2 + t] = logit[t];

    if (HAS_V) {
        float4 o = make_float4(0.f, 0.f, 0.f, 0.f);
#pragma unroll
        for (int t = 0; t < KS2; ++t) {
            if (nbr[t] >= 0) {  // wave-uniform
                const float4 vv = *(const float4*)(V + (size_t)nbr[t] * CCH + c0);
                o.x += logit[t] * vv.x;
                o.y += logit[t] * vv.y;
                o.z += logit[t] * vv.z;
                o.w += logit[t] * vv.w;
            }
        }
        *(float4*)(Out + (size_t)wid * CCH + c0) = o;
    }
}

// ---------------------------------------------------------------------------
// X = channels_norm(X + Vproj): mean / population-std over 128 channels,
// (x - m) / (std + 1e-5). One wave per position, shuffle reductions.
// ---------------------------------------------------------------------------
__global__ __launch_bounds__(256) void addnorm_kernel(
    const float* __restrict__ X, const float* __restrict__ Vp,
    float* __restrict__ Out)
{
    const int lane = threadIdx.x & 31;
    const int wid  = blockIdx.x * 8 + (threadIdx.x >> 5);
    if (wid >= NPOS) return;
    const int c0 = lane * 4;

    const float4 xv = *(const float4*)(X  + (size_t)wid * CCH + c0);
    const float4 vv = *(const float4*)(Vp + (size_t)wid * CCH + c0);
    float4 s;
    s.x = xv.x + vv.x; s.y = xv.y + vv.y; s.z = xv.z + vv.z; s.w = xv.w + vv.w;

    float sum = s.x + s.y + s.z + s.w;
#pragma unroll
    for (int off = 16; off >= 1; off >>= 1) sum += __shfl_xor(sum, off, 32);
    const float mean = sum * (1.0f / 128.0f);

    const float d0 = s.x - mean, d1 = s.y - mean, d2 = s.z - mean, d3 = s.w - mean;
    float vsum = d0 * d0 + d1 * d1 + d2 * d2 + d3 * d3;
#pragma unroll
    for (int off = 16; off >= 1; off >>= 1) vsum += __shfl_xor(vsum, off, 32);
    const float stdv = sqrtf(vsum * (1.0f / 128.0f));
    const float r = 1.0f / (stdv + 1e-5f);

    float4 o;
    o.x = d0 * r; o.y = d1 * r; o.z = d2 * r; o.w = d3 * r;
    *(float4*)(Out + (size_t)wid * CCH + c0) = o;
}

// ---------------------------------------------------------------------------
extern "C" void kernel_launch(void* const* d_in, const int* in_sizes, int n_in,
                              void* d_out, int out_size, void* d_ws, size_t ws_size,
                              hipStream_t stream)
{
    const float* x       = (const float*)d_in[0];
    const float* Wq_init = (const float*)d_in[1];
    const float* bq_init = (const float*)d_in[2];
    const float* Wk_init = (const float*)d_in[3];
    const float* bk_init = (const float*)d_in[4];
    const float* Wq      = (const float*)d_in[5];   // [2,128,128]
    const float* bq      = (const float*)d_in[6];   // [2,128]
    const float* Wk      = (const float*)d_in[7];
    const float* bk      = (const float*)d_in[8];
    const float* Wv      = (const float*)d_in[9];
    const float* bv      = (const float*)d_in[10];

    float* outX = (float*)d_out;                 // [32768,128]
    float* outM = outX + (size_t)NPOS * CCH;     // [32768,25]

    char* ws = (char*)d_ws;
    float* Qb    = (float*)ws; ws += (size_t)NPOS * CCH * sizeof(float);
    float* Kb    = (float*)ws; ws += (size_t)NPOS * CCH * sizeof(float);
    float* Vattn = (float*)ws; ws += (size_t)NPOS * CCH * sizeof(float);
    float* Xcur  = (float*)ws; ws += (size_t)NPOS * CCH * sizeof(float);
    float* Mbuf  = (float*)ws; ws += (size_t)NPOS * KS2 * sizeof(float);

    const dim3 gG(NPOS / 64), bG(128);   // gemm: 4 waves/block, 16 rows/wave
    const dim3 gA(NPOS / 8),  bA(256);   // attn/norm: 8 waves/block, 1 pos/wave

    // Initial scores-only attention (Q=K=x projected)
    gemm_wmma<<<gG, bG, 0, stream>>>(x, Wq_init, bq_init, Qb);
    gemm_wmma<<<gG, bG, 0, stream>>>(x, Wk_init, bk_init, Kb);
    attn_kernel<false, false><<<gA, bA, 0, stream>>>(Qb, Kb, nullptr, nullptr, nullptr, Mbuf);

    // Conv block 0: in = x, out = Xcur, M in-place in Mbuf
    gemm_wmma<<<gG, bG, 0, stream>>>(x, Wq + 0 * CCH * CCH, bq + 0 * CCH, Qb);
    gemm_wmma<<<gG, bG, 0, stream>>>(x, Wk + 0 * CCH * CCH, bk + 0 * CCH, Kb);
    attn_kernel<true, true><<<gA, bA, 0, stream>>>(Qb, Kb, x, Mbuf, Vattn, Mbuf);
    gemm_wmma<<<gG, bG, 0, stream>>>(Vattn, Wv + 0 * CCH * CCH, bv + 0 * CCH, Qb);
    addnorm_kernel<<<gA, bA, 0, stream>>>(x, Qb, Xcur);

    // Conv block 1: in = Xcur, final X/M straight into d_out
    gemm_wmma<<<gG, bG, 0, stream>>>(Xcur, Wq + 1 * CCH * CCH, bq + 1 * CCH, Qb);
    gemm_wmma<<<gG, bG, 0, stream>>>(Xcur, Wk + 1 * CCH * CCH, bk + 1 * CCH, Kb);
    attn_kernel<true, true><<<gA, bA, 0, stream>>>(Qb, Kb, Xcur, Mbuf, Vattn, outM);
    gemm_wmma<<<gG, bG, 0, stream>>>(Vattn, Wv + 1 * CCH * CCH, bv + 1 * CCH, Qb);
    addnorm_kernel<<<gA, bA, 0, stream>>>(Xcur, Qb, outX);
}